// DGITransductiveGConv_26345329393831
// MI455X (gfx1250) — compile-verified
//
#include <hip/hip_runtime.h>

typedef __attribute__((ext_vector_type(2))) float v2f;
typedef __attribute__((ext_vector_type(8))) float v8f;

#define N_NODES 50000
#define N_EDGES 800000
#define D 128

// ---------------- degree / norm precompute ----------------

__global__ void k_init_deg(float* __restrict__ deg) {
  int i = blockIdx.x * 256 + threadIdx.x;
  if (i < N_NODES) deg[i] = 1.0f;  // self-loop weight
}

__global__ void k_deg_accum(const int* __restrict__ dst,
                            const float* __restrict__ ew,
                            float* __restrict__ deg) {
  int e = blockIdx.x * 256 + threadIdx.x;
  if (e < N_EDGES) atomicAdd(&deg[dst[e]], ew[e]);
}

__global__ void k_deg_to_dinv(float* __restrict__ deg) {
  int i = blockIdx.x * 256 + threadIdx.x;
  if (i < N_NODES) {
    float d = deg[i];
    deg[i] = d > 0.0f ? rsqrtf(d) : 0.0f;
  }
}

// ---------------- dense GEMM: H = A * W via fp32 WMMA ----------------
// Block = 256 threads = 8 wave32s. Block computes rows [m0, m0+16) x all 128 cols.
// A-tile staged in LDS (row stride 132 -> conflict-free: 132 mod 64 = 4).
// Wave w computes the 16x16 tile at columns [16w, 16w+16) with 32 chained
// v_wmma_f32_16x16x4_f32 (K=128 in steps of 4).

__global__ __launch_bounds__(256)
void k_gemm_wmma(const float* __restrict__ A, const float* __restrict__ W,
                 float* __restrict__ H) {
  __shared__ float As[16 * 132];
  const int m0 = blockIdx.x * 16;

  // cooperative load of the 16x128 fp32 A tile (512 float4 chunks, 2 per thread)
  for (int c = threadIdx.x; c < 512; c += 256) {
    int row = c >> 5;
    int col = (c & 31) << 2;
    float4 v = *(const float4*)(A + (size_t)(m0 + row) * D + col);
    *(float4*)(&As[row * 132 + col]) = v;
  }
  __syncthreads();

  const int wave = threadIdx.x >> 5;
  const int lane = threadIdx.x & 31;
  const int lo   = lane & 15;            // M (for A) / N (for B) within tile
  const int hi   = (lane >> 4) << 1;     // K offset: 0 for lanes 0-15, 2 for 16-31
  const int n0   = wave << 4;

  v8f acc = {};
  for (int k0 = 0; k0 < D; k0 += 4) {
    // A frag: VGPR0 = K=k0+hi, VGPR1 = K=k0+hi+1 (contiguous -> one 8B LDS load)
    v2f a = *(const v2f*)(&As[lo * 132 + k0 + hi]);
    // B frag: row-striped, lanes 0-15 carry K=k0+{0,1}, lanes 16-31 K=k0+{2,3}
    v2f b;
    b.x = W[(size_t)(k0 + hi)     * D + n0 + lo];
    b.y = W[(size_t)(k0 + hi + 1) * D + n0 + lo];
    acc = __builtin_amdgcn_wmma_f32_16x16x4_f32(
        false, a, false, b, (short)0, acc, false, false);
  }

  // C/D layout: VGPR r -> lanes 0-15: (M=r, N=lane); lanes 16-31: (M=8+r, N=lane-16)
  const int rbase = m0 + ((lane >> 4) << 3);
#pragma unroll
  for (int r = 0; r < 8; ++r)
    H[(size_t)(rbase + r) * D + n0 + lo] = acc[r];
}

// ---------------- self-loop init: agg[i,:] = dinv[i]^2 * h[i,:] ----------------

__global__ void k_init_agg(const float* __restrict__ H,
                           const float* __restrict__ dinv,
                           float* __restrict__ agg) {
  size_t idx = (size_t)blockIdx.x * 256 + threadIdx.x;
  if (idx < (size_t)N_NODES * D) {
    int i = (int)(idx >> 7);
    float di = dinv[i];
    agg[idx] = di * di * H[idx];
  }
}

// ---------------- edge scatter: one wave per edge, float4 per lane ----------------

__global__ __launch_bounds__(256)
void k_scatter(const float* __restrict__ H, const int* __restrict__ src,
               const int* __restrict__ dst, const float* __restrict__ ew,
               const float* __restrict__ dinv, float* __restrict__ agg) {
  int e = blockIdx.x * 8 + (threadIdx.x >> 5);
  if (e >= N_EDGES) return;
  int lane = threadIdx.x & 31;
  int s = src[e];
  int d = dst[e];
  float w = dinv[s] * ew[e] * dinv[d];
  float4 v = ((const float4*)(H + (size_t)s * D))[lane];  // coalesced 512B row
  float* out = agg + (size_t)d * D + lane * 4;
  atomicAdd(out + 0, w * v.x);
  atomicAdd(out + 1, w * v.y);
  atomicAdd(out + 2, w * v.z);
  atomicAdd(out + 3, w * v.w);
}

// ---------------- bias + per-channel PReLU ----------------

__global__ void k_bias_prelu(const float* __restrict__ agg,
                             const float* __restrict__ b,
                             const float* __restrict__ a,
                             float* __restrict__ out) {
  size_t idx = (size_t)blockIdx.x * 256 + threadIdx.x;
  if (idx < (size_t)N_NODES * D) {
    int ch = (int)(idx & (D - 1));
    float z = agg[idx] + b[ch];
    out[idx] = z > 0.0f ? z : a[ch] * z;
  }
}

// ---------------- driver ----------------

extern "C" void kernel_launch(void* const* d_in, const int* in_sizes, int n_in,
                              void* d_out, int out_size, void* d_ws, size_t ws_size,
                              hipStream_t stream) {
  (void)in_sizes; (void)n_in; (void)out_size; (void)ws_size;

  const float* x   = (const float*)d_in[0];
  const int*   ei  = (const int*)d_in[1];
  const float* ew  = (const float*)d_in[2];
  const float* W1  = (const float*)d_in[3];
  const float* b1  = (const float*)d_in[4];
  const float* a1  = (const float*)d_in[5];
  const float* W2  = (const float*)d_in[6];
  const float* b2  = (const float*)d_in[7];
  const float* a2  = (const float*)d_in[8];
  const int* src = ei;
  const int* dst = ei + N_EDGES;

  float* zout = (float*)d_out;

  // workspace layout (floats): dinv[N] | h[N*D] | agg[N*D]
  float* dinv = (float*)d_ws;
  float* h    = dinv + N_NODES;
  float* agg  = h + (size_t)N_NODES * D;

  const int nodeBlocks = (N_NODES + 255) / 256;                    // 196
  const int edgeBlocks = (N_EDGES + 255) / 256;                    // 3125
  const int featBlocks = (int)(((size_t)N_NODES * D + 255) / 256); // 25000
  const int gemmBlocks = N_NODES / 16;                             // 3125
  const int scatBlocks = N_EDGES / 8;                              // 100000

  // symmetric normalization coefficients
  k_init_deg<<<nodeBlocks, 256, 0, stream>>>(dinv);
  k_deg_accum<<<edgeBlocks, 256, 0, stream>>>(dst, ew, dinv);
  k_deg_to_dinv<<<nodeBlocks, 256, 0, stream>>>(dinv);

  // ---- layer 1: z1 = prelu(D^-1/2 (A+I) D^-1/2 (x W1) + b1, a1) -> agg ----
  k_gemm_wmma<<<gemmBlocks, 256, 0, stream>>>(x, W1, h);
  k_init_agg<<<featBlocks, 256, 0, stream>>>(h, dinv, agg);
  k_scatter<<<scatBlocks, 256, 0, stream>>>(h, src, dst, ew, dinv, agg);
  k_bias_prelu<<<featBlocks, 256, 0, stream>>>(agg, b1, a1, agg);  // in place -> z1

  // ---- layer 2: out = prelu(D^-1/2 (A+I) D^-1/2 (z1 W2) + b2, a2) ----
  k_gemm_wmma<<<gemmBlocks, 256, 0, stream>>>(agg, W2, h);
  k_init_agg<<<featBlocks, 256, 0, stream>>>(h, dinv, zout);       // also fully overwrites d_out
  k_scatter<<<scatBlocks, 256, 0, stream>>>(h, src, dst, ew, dinv, zout);
  k_bias_prelu<<<featBlocks, 256, 0, stream>>>(zout, b2, a2, zout);
}